// LinkPredictionModel_34411277975925
// MI455X (gfx1250) — compile-verified
//
#include <hip/hip_runtime.h>
#include <hip/hip_bf16.h>

// GNN (NNConv-like) forward, restructured so the scatter-mean commutes past the
// edge GEMM:  aggr = (segsum(h[src]) @ Wn_top + segsum(ea) @ Wn_bot)/max(cnt,1) + bn*(cnt>0)
// Edge work = raw-feature atomic scatter (L2-resident on MI455X's 192MB L2);
// node GEMMs use V_WMMA_F32_16X16X4_F32 with weights staged transposed in LDS
// (B fragments become single ds_load_b64 ops -> no per-WMMA global-load stalls).

#define N_NODES 100000
#define BN_EPS 1e-5f

typedef __attribute__((ext_vector_type(2))) float v2f;
typedef __attribute__((ext_vector_type(8))) float v8f;

// ---------------------------------------------------------------- zero
__global__ void zero_f32_kernel(float* __restrict__ p, int n) {
  int i = blockIdx.x * blockDim.x + threadIdx.x;
  int stride = gridDim.x * blockDim.x;
  for (; i < n; i += stride) p[i] = 0.0f;
}

// ---------------------------------------------------------------- edge_attr scatter (once; layer-invariant)
// 4 threads per edge, 4 channels each (float4 load, 4 L2 atomics).
__global__ void scatter_ea_kernel(const int* __restrict__ ei, const float* __restrict__ ea,
                                  float* __restrict__ aggea, float* __restrict__ cnt, int E) {
  int gid = blockIdx.x * blockDim.x + threadIdx.x;
  if (gid >= E * 4) return;
  int e  = gid >> 2;
  int c4 = (gid & 3) << 2;
  int d  = ei[E + e];  // dst
  float4 v = *(const float4*)&ea[e * 16 + c4];
  float* p = &aggea[d * 16 + c4];
  atomicAdd(p + 0, v.x); atomicAdd(p + 1, v.y);
  atomicAdd(p + 2, v.z); atomicAdd(p + 3, v.w);
  if ((gid & 3) == 0) atomicAdd(&cnt[d], 1.0f);
}

// ---------------------------------------------------------------- h[src] scatter per layer
// 16 threads per edge, 4 channels each. h and aggx are L2-resident (25MB each).
__global__ void scatter_h_kernel(const int* __restrict__ ei, const float* __restrict__ h,
                                 float* __restrict__ aggx, int E) {
  int gid = blockIdx.x * blockDim.x + threadIdx.x;
  if (gid >= E * 16) return;
  int e  = gid >> 4;
  int c4 = (gid & 15) << 2;
  int s  = ei[e];
  int d  = ei[E + e];
  float4 v = *(const float4*)&h[s * 64 + c4];
  float* p = &aggx[d * 64 + c4];
  atomicAdd(p + 0, v.x); atomicAdd(p + 1, v.y);
  atomicAdd(p + 2, v.z); atomicAdd(p + 3, v.w);
}

// ---------------------------------------------------------------- fused node update (WMMA f32)
// One wave per 16-node block; DOUT/16 column tiles. Weights staged in LDS,
// transposed: sWn[col][k] (k = 0..79), sWs[col][k] (k = 0..63) so each B
// fragment (k, k+1 for one column) is one aligned 8-byte LDS load.
// Computes act = relu( h@Ws + bs + (aggx@Wn[0:64] + aggea@Wn[64:80])/max(cnt,1) + bn*(cnt>0) ),
// stores act (pre-BN) and accumulates per-channel sum / sumsq for batchnorm.
template <int DOUT>
__global__ void node_update_kernel(const float* __restrict__ h,      // [N,64]
                                   const float* __restrict__ aggx,   // [N,64]
                                   const float* __restrict__ aggea,  // [N,16]
                                   const float* __restrict__ cnt,    // [N]
                                   const float* __restrict__ Wn,     // [80,DOUT]
                                   const float* __restrict__ bn,     // [DOUT]
                                   const float* __restrict__ Ws,     // [64,DOUT]
                                   const float* __restrict__ bs,     // [DOUT]
                                   float* __restrict__ act,          // [N,DOUT]
                                   float* __restrict__ stats) {      // [2*DOUT] sum, sumsq
  __shared__ float sWn[DOUT * 80];   // transposed: [col][k]
  __shared__ float sWs[DOUT * 64];   // transposed: [col][k]

  // cooperative staging (coalesced global reads, transposed LDS stores)
  for (int idx = threadIdx.x; idx < 80 * DOUT; idx += blockDim.x) {
    int k = idx / DOUT, col = idx & (DOUT - 1);
    sWn[col * 80 + k] = Wn[idx];
  }
  for (int idx = threadIdx.x; idx < 64 * DOUT; idx += blockDim.x) {
    int k = idx / DOUT, col = idx & (DOUT - 1);
    sWs[col * 64 + k] = Ws[idx];
  }
  __syncthreads();

  const int lane = threadIdx.x & 31;
  const int wave = threadIdx.x >> 5;
  const int wid  = blockIdx.x * (blockDim.x >> 5) + wave;
  const int m0   = wid * 16;
  if (m0 >= N_NODES) return;                  // wave-uniform: EXEC stays all-ones for WMMA
  const int half = lane >> 4;                 // 0: K=0,1 / C rows 0..7 ; 1: K=2,3 / rows 8..15
  const int l15  = lane & 15;
  const int koff = half * 2;
  constexpr int NT = DOUT / 16;

  v8f accN[NT];                               // neighbor path
  v8f accS[NT];                               // self path
#pragma unroll
  for (int t = 0; t < NT; ++t) { accN[t] = (v8f)0.0f; accS[t] = (v8f)0.0f; }

  // aggx @ Wn_top  (K = 64)
#pragma unroll
  for (int k0 = 0; k0 < 64; k0 += 4) {
    v2f a = *(const v2f*)&aggx[(m0 + l15) * 64 + k0 + koff];
#pragma unroll
    for (int t = 0; t < NT; ++t) {
      v2f b = *(const v2f*)&sWn[(t * 16 + l15) * 80 + k0 + koff];
      accN[t] = __builtin_amdgcn_wmma_f32_16x16x4_f32(false, a, false, b, (short)0,
                                                      accN[t], false, false);
    }
  }
  // aggea @ Wn_bot  (K = 16, Wn rows 64..79)
#pragma unroll
  for (int k0 = 0; k0 < 16; k0 += 4) {
    v2f a = *(const v2f*)&aggea[(m0 + l15) * 16 + k0 + koff];
#pragma unroll
    for (int t = 0; t < NT; ++t) {
      v2f b = *(const v2f*)&sWn[(t * 16 + l15) * 80 + 64 + k0 + koff];
      accN[t] = __builtin_amdgcn_wmma_f32_16x16x4_f32(false, a, false, b, (short)0,
                                                      accN[t], false, false);
    }
  }
  // h @ Ws  (K = 64)
#pragma unroll
  for (int k0 = 0; k0 < 64; k0 += 4) {
    v2f a = *(const v2f*)&h[(m0 + l15) * 64 + k0 + koff];
#pragma unroll
    for (int t = 0; t < NT; ++t) {
      v2f b = *(const v2f*)&sWs[(t * 16 + l15) * 64 + k0 + koff];
      accS[t] = __builtin_amdgcn_wmma_f32_16x16x4_f32(false, a, false, b, (short)0,
                                                      accS[t], false, false);
    }
  }

  // per-row 1/max(cnt,1) and cnt>0 mask; C-tile row r maps to node m0 + r + half*8
  float invc[8], msk[8];
#pragma unroll
  for (int r = 0; r < 8; ++r) {
    float c = cnt[m0 + r + half * 8];
    invc[r] = 1.0f / fmaxf(c, 1.0f);
    msk[r]  = (c > 0.0f) ? 1.0f : 0.0f;
  }

#pragma unroll
  for (int t = 0; t < NT; ++t) {
    int col = t * 16 + l15;
    float bnc = bn[col];
    float bsc = bs[col];
    float s = 0.0f, ss = 0.0f;
#pragma unroll
    for (int r = 0; r < 8; ++r) {
      float v = accN[t][r] * invc[r] + bnc * msk[r] + accS[t][r] + bsc;
      v = fmaxf(v, 0.0f);
      act[(m0 + r + half * 8) * DOUT + col] = v;
      s += v;
      ss += v * v;
    }
    atomicAdd(&stats[col], s);
    atomicAdd(&stats[DOUT + col], ss);
  }
}

// ---------------------------------------------------------------- batchnorm apply (+optional relu)
template <int DOUT, bool RELU>
__global__ void bn_apply_kernel(const float* __restrict__ act, const float* __restrict__ stats,
                                const float* __restrict__ gamma, const float* __restrict__ beta,
                                float* __restrict__ out) {
  int gid = blockIdx.x * blockDim.x + threadIdx.x;
  if (gid >= N_NODES * DOUT) return;
  int col = gid & (DOUT - 1);
  const float invN = 1.0f / (float)N_NODES;
  float mean = stats[col] * invN;
  float var  = stats[DOUT + col] * invN - mean * mean;  // biased, as reference
  float v = (act[gid] - mean) * rsqrtf(var + BN_EPS) * gamma[col] + beta[col];
  if (RELU) v = fmaxf(v, 0.0f);
  out[gid] = v;
}

// ---------------------------------------------------------------- launch
extern "C" void kernel_launch(void* const* d_in, const int* in_sizes, int n_in,
                              void* d_out, int out_size, void* d_ws, size_t ws_size,
                              hipStream_t stream) {
  const float* x  = (const float*)d_in[0];
  const int*   ei = (const int*)d_in[1];   // [2,E]
  const float* ea = (const float*)d_in[2];
  const int E = in_sizes[1] / 2;

  // params: 3 layers x (Wn, bn, Ws, bs, gamma, beta) starting at d_in[3]
  const float* P[18];
  for (int i = 0; i < 18; ++i) P[i] = (const float*)d_in[3 + i];

  // workspace layout (floats)
  float* ws    = (float*)d_ws;
  float* aggea = ws;                         // N*16
  float* cnt   = aggea + N_NODES * 16;       // N      (contiguous with aggea for one zero pass)
  float* aggx  = cnt + N_NODES;              // N*64
  float* stats = aggx + N_NODES * 64;        // 128    (contiguous with aggx for one zero pass)
  float* hcur  = stats + 128;                // N*64
  float* hp    = hcur + N_NODES * 64;        // N*64
  (void)ws_size; (void)n_in; (void)out_size;

  const int ZB = 2048, TPB = 256;

  // edge_attr aggregation + degree counts: layer-invariant, done once per call
  zero_f32_kernel<<<ZB, TPB, 0, stream>>>(aggea, N_NODES * 17);
  scatter_ea_kernel<<<(E * 4 + TPB - 1) / TPB, TPB, 0, stream>>>(ei, ea, aggea, cnt, E);

  const int nodeBlocks = ((N_NODES / 16) + 7) / 8;  // 8 waves (wave32) per 256-thread block

  const float* hin = x;
  for (int L = 0; L < 3; ++L) {
    const float* Wn = P[6 * L + 0];
    const float* bn = P[6 * L + 1];
    const float* Ws = P[6 * L + 2];
    const float* bs = P[6 * L + 3];
    const float* gm = P[6 * L + 4];
    const float* bt = P[6 * L + 5];

    zero_f32_kernel<<<ZB, TPB, 0, stream>>>(aggx, N_NODES * 64 + 128);  // aggx + stats
    scatter_h_kernel<<<(E * 16 + TPB - 1) / TPB, TPB, 0, stream>>>(ei, hin, aggx, E);

    if (L < 2) {
      node_update_kernel<64><<<nodeBlocks, TPB, 0, stream>>>(hin, aggx, aggea, cnt,
                                                             Wn, bn, Ws, bs, hp, stats);
      bn_apply_kernel<64, true><<<(N_NODES * 64 + TPB - 1) / TPB, TPB, 0, stream>>>(
          hp, stats, gm, bt, hcur);
      hin = hcur;
    } else {
      node_update_kernel<32><<<nodeBlocks, TPB, 0, stream>>>(hin, aggx, aggea, cnt,
                                                             Wn, bn, Ws, bs, hp, stats);
      bn_apply_kernel<32, false><<<(N_NODES * 32 + TPB - 1) / TPB, TPB, 0, stream>>>(
          hp, stats, gm, bt, (float*)d_out);
    }
  }
}